// TGN_49959059587648
// MI455X (gfx1250) — compile-verified
//
#include <hip/hip_runtime.h>
#include <hip/hip_bf16.h>
#include <math.h>

#if defined(__HIP_DEVICE_COMPILE__) && !__has_builtin(__builtin_amdgcn_wmma_f32_16x16x4_f32)
#error "gfx1250 fp32 WMMA builtin not available on this toolchain"
#endif

typedef __attribute__((ext_vector_type(2))) float v2f;
typedef __attribute__((ext_vector_type(8))) float v8f;

#define NNODES 100000
#define NEDGES 500000
#define DMEM   128
#define EFD    16
#define HOUT   256   // heads * D
#define OUTC   64

// ---------------- WMMA helpers (V_WMMA_F32_16X16X4_F32) -------------------
// A: 16x4 fp32 -> 2 VGPRs/lane : lane L holds A[L&15][k0 + 2*(L>>4) + {0,1}]
// B: 4x16 fp32 -> 2 VGPRs/lane : lane L holds B[k0 + 2*(L>>4) + {0,1}][L&15]
// C/D: 16x16 fp32, 8 VGPRs/lane: elem r  <-> row (r + 8*(L>>4)), col (L&15)

__device__ __forceinline__ v8f wmma4(v2f a, v2f b, v8f c) {
  return __builtin_amdgcn_wmma_f32_16x16x4_f32(
      /*neg_a=*/false, a, /*neg_b=*/false, b,
      /*c_mod=*/(short)0, c, /*reuse_a=*/false, /*reuse_b=*/false);
}

__device__ __forceinline__ v2f ldA(const float* base, int lda, int k0) {
  int lane = threadIdx.x & 31;
  const float* p = base + (lane & 15) * lda + k0 + ((lane >> 4) << 1);
  v2f a; a.x = p[0]; a.y = p[1]; return a;
}

// B[k][n] = W[k*ldb + n]   (row-major weight, X @ W)
__device__ __forceinline__ v2f ldB(const float* W, int ldb, int k0, int n0) {
  int lane = threadIdx.x & 31;
  int col = n0 + (lane & 15);
  int kk  = k0 + ((lane >> 4) << 1);
  v2f b; b.x = W[(size_t)kk * ldb + col]; b.y = W[(size_t)(kk + 1) * ldb + col];
  return b;
}

// B[k][c] = W[c*ldw + k]   (for X @ W^T, W row-major [C,K])
__device__ __forceinline__ v2f ldBT(const float* W, int ldw, int k0, int c0) {
  int lane = threadIdx.x & 31;
  const float* p = W + (size_t)(c0 + (lane & 15)) * ldw + k0 + ((lane >> 4) << 1);
  v2f b; b.x = p[0]; b.y = p[1]; return b;
}

// order-preserving float<->uint for atomic max
__device__ __forceinline__ unsigned f2o(float f) {
  unsigned u = __float_as_uint(f);
  return (u & 0x80000000u) ? ~u : (u | 0x80000000u);
}
__device__ __forceinline__ float o2f(unsigned u) {
  u = (u & 0x80000000u) ? (u & 0x7fffffffu) : ~u;
  return __uint_as_float(u);
}

// ---------------- node_mem = copy(memory) ---------------------------------
__global__ void k_copy(const float* __restrict__ mem, float* __restrict__ node_mem) {
  size_t t = (size_t)blockIdx.x * blockDim.x + threadIdx.x;
  if (t * 4 < (size_t)NNODES * DMEM)
    *(float4*)(node_mem + t * 4) = *(const float4*)(mem + t * 4);
}

// ---------------- fused edge kernel: MLP message + GRU + scatter ----------
__global__ __launch_bounds__(32)
void k_edge(const int* __restrict__ ei, const float* __restrict__ eattr,
            const float* __restrict__ mem,
            const float* __restrict__ w1, const float* __restrict__ b1,
            const float* __restrict__ w2, const float* __restrict__ b2,
            const float* __restrict__ wih, const float* __restrict__ whh,
            const float* __restrict__ bih, const float* __restrict__ bhh,
            float* __restrict__ node_mem)
{
  __shared__ float sA[16 * 272];   // [src_mem | dst_mem | edge_attr], lda=272
  __shared__ float sH[16 * 128];   // relu(h1)
  __shared__ float sG[16 * 128];   // messages
  __shared__ int   sDst[16];

  const int lane = threadIdx.x;
  const int e0   = blockIdx.x * 16;
  const int* src = ei;
  const int* dst = ei + NEDGES;

  if (lane < 16) sDst[lane] = dst[e0 + lane];
  for (int t = lane; t < 16 * 128; t += 32) {
    int r = t >> 7, c = t & 127;
    sA[r * 272 + c] = mem[(size_t)src[e0 + r] * DMEM + c];
  }
  for (int t = lane; t < 16 * 128; t += 32) {
    int r = t >> 7, c = t & 127;
    sA[r * 272 + 128 + c] = mem[(size_t)dst[e0 + r] * DMEM + c];
  }
  for (int t = lane; t < 16 * 16; t += 32) {
    int r = t >> 4, c = t & 15;
    sA[r * 272 + 256 + c] = eattr[(size_t)(e0 + r) * EFD + c];
  }
  __syncthreads();

  // h1 = relu(msg_in @ w1 + b1)      [16,272]x[272,128]
  for (int nt = 0; nt < 8; ++nt) {
    int n0 = nt * 16;
    v8f acc = {};
    for (int k0 = 0; k0 < 272; k0 += 4)
      acc = wmma4(ldA(sA, 272, k0), ldB(w1, 128, k0, n0), acc);
    float bias = b1[n0 + (lane & 15)];
#pragma unroll
    for (int r = 0; r < 8; ++r) {
      int row = r + ((lane >> 4) << 3);
      float v = acc[r] + bias;
      sH[row * 128 + n0 + (lane & 15)] = v > 0.f ? v : 0.f;
    }
  }
  __syncthreads();

  // messages = h1 @ w2 + b2          [16,128]x[128,128]
  for (int nt = 0; nt < 8; ++nt) {
    int n0 = nt * 16;
    v8f acc = {};
    for (int k0 = 0; k0 < 128; k0 += 4)
      acc = wmma4(ldA(sH, 128, k0), ldB(w2, 128, k0, n0), acc);
    float bias = b2[n0 + (lane & 15)];
#pragma unroll
    for (int r = 0; r < 8; ++r) {
      int row = r + ((lane >> 4) << 3);
      sG[row * 128 + n0 + (lane & 15)] = acc[r] + bias;
    }
  }
  __syncthreads();

  // GRU: gi = msg @ wih^T, gh = dst_mem @ whh^T ; gates per 16-col block
  for (int j = 0; j < 8; ++j) {
    int c0 = j * 16;
    v8f ri = {}, rh = {}, zi = {}, zh = {}, ni = {}, nh = {};
    for (int k0 = 0; k0 < 128; k0 += 4) {
      v2f am = ldA(sG, 128, k0);        // messages tile
      v2f ad = ldA(sA + 128, 272, k0);  // dst_mem tile
      ri = wmma4(am, ldBT(wih, 128, k0, c0),       ri);
      zi = wmma4(am, ldBT(wih, 128, k0, c0 + 128), zi);
      ni = wmma4(am, ldBT(wih, 128, k0, c0 + 256), ni);
      rh = wmma4(ad, ldBT(whh, 128, k0, c0),       rh);
      zh = wmma4(ad, ldBT(whh, 128, k0, c0 + 128), zh);
      nh = wmma4(ad, ldBT(whh, 128, k0, c0 + 256), nh);
    }
    int c = c0 + (lane & 15);
    float birv = bih[c], bizv = bih[c + 128], binv = bih[c + 256];
    float bhrv = bhh[c], bhzv = bhh[c + 128], bhnv = bhh[c + 256];
#pragma unroll
    for (int r = 0; r < 8; ++r) {
      int row = r + ((lane >> 4) << 3);
      float rg = 1.f / (1.f + __expf(-(ri[r] + birv + rh[r] + bhrv)));
      float zg = 1.f / (1.f + __expf(-(zi[r] + bizv + zh[r] + bhzv)));
      float ng = tanhf(ni[r] + binv + rg * (nh[r] + bhnv));
      float dm = sA[row * 272 + 128 + c];
      node_mem[(size_t)sDst[row] * DMEM + c] = (1.f - zg) * ng + zg * dm;
    }
  }
}

// ---------------- Q/K/V/skip projections ----------------------------------
__device__ __forceinline__ void gemm16_row(const float* sA, const float* W,
                                           const float* bias, float* out,
                                           size_t row0) {
  const int lane = threadIdx.x & 31;
  for (int n0 = 0; n0 < HOUT; n0 += 16) {
    v8f acc = {};
    for (int k0 = 0; k0 < 128; k0 += 4)
      acc = wmma4(ldA(sA, 128, k0), ldB(W, HOUT, k0, n0), acc);
    float b = bias[n0 + (lane & 15)];
#pragma unroll
    for (int r = 0; r < 8; ++r) {
      size_t row = row0 + r + ((lane >> 4) << 3);
      out[row * HOUT + n0 + (lane & 15)] = acc[r] + b;
    }
  }
}

__global__ __launch_bounds__(32)
void k_qkvs(const float* __restrict__ node_mem,
            const float* wq, const float* bq, const float* wk, const float* bk,
            const float* wv, const float* bv, const float* wsk, const float* bsk,
            float* qo, float* ko, float* vo, float* co)
{
  __shared__ float sA[16 * 128];
  const int lane = threadIdx.x;
  size_t row0 = (size_t)blockIdx.x * 16;
  for (int t = lane; t < 16 * 128; t += 32)
    sA[t] = node_mem[row0 * DMEM + t];     // lda==128, tile is contiguous
  __syncthreads();
  gemm16_row(sA, wq,  bq,  qo, row0);
  gemm16_row(sA, wk,  bk,  ko, row0);
  gemm16_row(sA, wv,  bv,  vo, row0);
  gemm16_row(sA, wsk, bsk, co, row0);      // conv buffer initialized with skip
}

// ---------------- attention: logits + running max -------------------------
__global__ __launch_bounds__(256)
void k_logits(const int* __restrict__ ei, const float* __restrict__ qb,
              const float* __restrict__ kb, float* __restrict__ logits,
              unsigned* __restrict__ m_enc)
{
  int wave = (int)(((size_t)blockIdx.x * blockDim.x + threadIdx.x) >> 5);
  int lane = threadIdx.x & 31;
  if (wave >= NEDGES) return;
  int s = ei[wave];
  int d = ei[NEDGES + wave];
  const float* qrow = qb + (size_t)d * HOUT;
  const float* krow = kb + (size_t)s * HOUT;
#pragma unroll
  for (int h = 0; h < 2; ++h) {
    float4 qv = *(const float4*)(qrow + h * 128 + lane * 4);
    float4 kv = *(const float4*)(krow + h * 128 + lane * 4);
    float v = qv.x * kv.x + qv.y * kv.y + qv.z * kv.z + qv.w * kv.w;
#pragma unroll
    for (int off = 16; off >= 1; off >>= 1) v += __shfl_xor(v, off, 32);
    if (lane == 0) {
      v *= 0.0883883476483184f;   // 1/sqrt(128)
      logits[(size_t)wave * 2 + h] = v;
      atomicMax(&m_enc[(size_t)d * 2 + h], f2o(v));
    }
  }
}

// ---------------- softmax: exp + segment sum ------------------------------
__global__ void k_soft(const int* __restrict__ ei, float* __restrict__ logits,
                       const unsigned* __restrict__ m_enc, float* __restrict__ ssum)
{
  int t = blockIdx.x * blockDim.x + threadIdx.x;
  if (t >= 2 * NEDGES) return;
  int e = t >> 1, h = t & 1;
  int d = ei[NEDGES + e];
  float ex = __expf(logits[t] - o2f(m_enc[(size_t)d * 2 + h]));
  logits[t] = ex;                          // reuse buffer for ex
  atomicAdd(&ssum[(size_t)d * 2 + h], ex);
}

// ---------------- weighted value aggregation ------------------------------
__global__ void k_agg(const int* __restrict__ ei, const float* __restrict__ ex,
                      const float* __restrict__ ssum, const float* __restrict__ vb,
                      float* __restrict__ convb)
{
  long long t = (long long)blockIdx.x * blockDim.x + threadIdx.x;
  if (t >= (long long)NEDGES * 64) return;
  int e  = (int)(t >> 6);
  int d0 = (int)(t & 63) * 4;
  int h  = d0 >> 7;
  int s = ei[e];
  int d = ei[NEDGES + e];
  float alpha = ex[(size_t)e * 2 + h] / (ssum[(size_t)d * 2 + h] + 1e-16f);
  float4 vv = *(const float4*)(vb + (size_t)s * HOUT + d0);
  float* o = convb + (size_t)d * HOUT + d0;
  atomicAdd(o + 0, vv.x * alpha);
  atomicAdd(o + 1, vv.y * alpha);
  atomicAdd(o + 2, vv.z * alpha);
  atomicAdd(o + 3, vv.w * alpha);
}

// ---------------- classifier ----------------------------------------------
__global__ __launch_bounds__(32)
void k_cls(const float* __restrict__ convb, const float* __restrict__ wc,
           const float* __restrict__ bc, float* __restrict__ out)
{
  __shared__ float sA[16 * 256];
  const int lane = threadIdx.x;
  size_t row0 = (size_t)blockIdx.x * 16;
  for (int t = lane; t < 16 * 256; t += 32)
    sA[t] = convb[row0 * HOUT + t];        // lda==256, contiguous
  __syncthreads();
  for (int n0 = 0; n0 < OUTC; n0 += 16) {
    v8f acc = {};
    for (int k0 = 0; k0 < 256; k0 += 4)
      acc = wmma4(ldA(sA, 256, k0), ldB(wc, OUTC, k0, n0), acc);
    float b = bc[n0 + (lane & 15)];
#pragma unroll
    for (int r = 0; r < 8; ++r) {
      size_t row = row0 + r + ((lane >> 4) << 3);
      out[row * OUTC + n0 + (lane & 15)] = acc[r] + b;
    }
  }
}

// ---------------- launch ---------------------------------------------------
extern "C" void kernel_launch(void* const* d_in, const int* in_sizes, int n_in,
                              void* d_out, int out_size, void* d_ws, size_t ws_size,
                              hipStream_t stream)
{
  const int*   ei    = (const int*)d_in[0];
  const float* eattr = (const float*)d_in[1];
  /* d_in[2] = edge_time: unused by reference */
  const float* mem = (const float*)d_in[3];
  const float* w1  = (const float*)d_in[4];  const float* b1  = (const float*)d_in[5];
  const float* w2  = (const float*)d_in[6];  const float* b2  = (const float*)d_in[7];
  const float* wih = (const float*)d_in[8];  const float* whh = (const float*)d_in[9];
  const float* bih = (const float*)d_in[10]; const float* bhh = (const float*)d_in[11];
  const float* wq  = (const float*)d_in[12]; const float* bq  = (const float*)d_in[13];
  const float* wk  = (const float*)d_in[14]; const float* bk  = (const float*)d_in[15];
  const float* wv  = (const float*)d_in[16]; const float* bv  = (const float*)d_in[17];
  const float* wsk = (const float*)d_in[18]; const float* bsk = (const float*)d_in[19];
  const float* wc  = (const float*)d_in[20]; const float* bc  = (const float*)d_in[21];
  float* out = (float*)d_out;

  // workspace layout (fp32 elements)
  float* ws       = (float*)d_ws;
  float* node_mem = ws;                                   // N*128
  float* qb       = node_mem + (size_t)NNODES * DMEM;     // N*256
  float* kb       = qb  + (size_t)NNODES * HOUT;          // N*256
  float* vb       = kb  + (size_t)NNODES * HOUT;          // N*256
  float* convb    = vb  + (size_t)NNODES * HOUT;          // N*256
  float* logitsb  = convb + (size_t)NNODES * HOUT;        // 2E (logits -> ex)
  unsigned* m_enc = (unsigned*)(logitsb + (size_t)NEDGES * 2);  // 2N
  float* ssum     = (float*)(m_enc + (size_t)NNODES * 2);       // 2N

  (void)hipMemsetAsync(m_enc, 0, (size_t)NNODES * 2 * sizeof(unsigned), stream);
  (void)hipMemsetAsync(ssum,  0, (size_t)NNODES * 2 * sizeof(float), stream);

  k_copy<<<(NNODES * DMEM / 4 + 255) / 256, 256, 0, stream>>>(mem, node_mem);
  k_edge<<<NEDGES / 16, 32, 0, stream>>>(ei, eattr, mem, w1, b1, w2, b2,
                                         wih, whh, bih, bhh, node_mem);
  k_qkvs<<<NNODES / 16, 32, 0, stream>>>(node_mem, wq, bq, wk, bk, wv, bv,
                                         wsk, bsk, qb, kb, vb, convb);
  k_logits<<<(NEDGES * 32) / 256, 256, 0, stream>>>(ei, qb, kb, logitsb, m_enc);
  k_soft<<<(2 * NEDGES + 255) / 256, 256, 0, stream>>>(ei, logitsb, m_enc, ssum);
  k_agg<<<(int)(((long long)NEDGES * 64 + 255) / 256), 256, 0, stream>>>(
      ei, logitsb, ssum, vb, convb);
  k_cls<<<NNODES / 16, 32, 0, stream>>>(convb, wc, bc, out);
}